// VectorQuantizer_36807869727044
// MI455X (gfx1250) — compile-verified
//
#include <hip/hip_runtime.h>
#include <math.h>

typedef __attribute__((ext_vector_type(2))) float v2f;
typedef __attribute__((ext_vector_type(8))) float v8f;
typedef __attribute__((ext_vector_type(4))) int   v4i;

#define NPTS    65536          // 64*32*32 points
#define DIM     256
#define KCODES  1024
#define BM      32             // points per block
#define CHUNK   128            // codebook rows per LDS chunk (CDNA5 320KB LDS makes this fit)
#define NCHUNK  (KCODES / CHUNK)
#define LDA     260            // padded row stride (floats) for A/B tiles (conflict-free WMMA fetch)
#define LDD     1042           // padded row stride (floats) for dist tile (conflict-free store+scan)
#define QOFF    (NPTS * DIM)   // 16777216 : offset of loss in d_out; indices follow at QOFF+1
#define INV_T   100.0f         // 1 / ENTROPY_TEMPERATURE

// workspace float layout
#define WS_AVG  0              // [0, 1024)    : sum over points of softmax probs
#define WS_ENT  1024           // sum over points of (T/S - lnS) = sum of sum_k p*logp
#define WS_MSE  1025           // sum of (q-x)^2
#define WS_C2   1026           // [1026, 2050) : |codebook_k|^2

#define GLOBAL_AS __attribute__((address_space(1)))
#define LDS_AS    __attribute__((address_space(3)))

#if __has_builtin(__builtin_amdgcn_global_load_async_to_lds_b128)
#define ASYNC_LDS 1
#define ASYNC_COPY_B128(g, l)                                                   \
    __builtin_amdgcn_global_load_async_to_lds_b128(                             \
        (GLOBAL_AS v4i*)(GLOBAL_AS void*)(void*)(g),                            \
        (LDS_AS v4i*)(LDS_AS void*)(void*)(l), 0, 0)
#else
#define ASYNC_LDS 0
#endif

// ---------------------------------------------------------------- zero accumulators
__global__ void vq_zero_kernel(float* __restrict__ ws) {
    for (int i = threadIdx.x; i < 1026; i += 256) ws[i] = 0.0f;
}

// ---------------------------------------------------------------- codebook row norms
__global__ __launch_bounds__(256)
void vq_c2_kernel(const float* __restrict__ cb, float* __restrict__ ws) {
    int k = blockIdx.x * 256 + threadIdx.x;
    if (k >= KCODES) return;
    const float4* r = (const float4*)(cb + (size_t)k * DIM);
    float s = 0.0f;
#pragma unroll 8
    for (int i = 0; i < DIM / 4; ++i) {
        float4 v = r[i];
        s += v.x * v.x + v.y * v.y + v.z * v.z + v.w * v.w;
    }
    ws[WS_C2 + k] = s;
}

// ---------------------------------------------------------------- fused distances + argmin + entropy stats
__global__ __launch_bounds__(256)
void vq_dist_kernel(const float* __restrict__ x, const float* __restrict__ cb,
                    float* __restrict__ out, float* __restrict__ ws) {
    extern __shared__ float smem[];
    float* As = smem;                   // 32  x 260
    float* Bs = As + BM * LDA;          // 128 x 260
    float* Ds = Bs + CHUNK * LDA;       // 32  x 1042  (relative distances, all 1024 codes)
    float* Pp = Ds + BM * LDD;          // 1024        (per-block avg_probs staging)

    const int tid   = threadIdx.x;
    const int pbase = blockIdx.x * BM;

    for (int j = tid; j < KCODES; j += 256) Pp[j] = 0.0f;

    // ---- stage A tile (32 points x 256 dims) : async memory->LDS on CDNA5
#if ASYNC_LDS
#pragma unroll
    for (int i = 0; i < (BM * DIM / 4) / 256; ++i) {      // 8 iters
        int f   = i * 256 + tid;
        int row = f >> 6;
        int c4  = f & 63;
        const float* g = x + (size_t)(pbase + row) * DIM + c4 * 4;
        float*       l = &As[row * LDA + c4 * 4];
        ASYNC_COPY_B128(g, l);
    }
#else
    const float4* x4 = (const float4*)x + (size_t)pbase * (DIM / 4);
#pragma unroll
    for (int i = 0; i < (BM * DIM / 4) / 256; ++i) {      // 8 iters
        int f   = i * 256 + tid;
        int row = f >> 6;
        int c4  = f & 63;
        float4 v = x4[row * (DIM / 4) + c4];
        *(float4*)&As[row * LDA + c4 * 4] = v;
    }
#endif

    const int wave   = tid >> 5;
    const int lane   = tid & 31;
    const int lane16 = lane & 15;
    const int koff   = (lane < 16) ? 0 : 2;               // A/B 16x4 fp32 operand layout (ISA 7.12.2)
    const int mtile  = wave >> 2;                         // 2 M-tiles of 16 points
    const int npair  = wave & 3;                          // 4 N-pairs of 32 codes per chunk
    const float* aRow  = &As[(mtile * 16 + lane16) * LDA];
    const float* bRow0 = &Bs[(npair * 32 + lane16) * LDA];
    const float* bRow1 = bRow0 + 16 * LDA;

#if ASYNC_LDS
    asm volatile("s_wait_asynccnt 0x0" ::: "memory");
#endif
    __syncthreads();

    for (int ch = 0; ch < NCHUNK; ++ch) {
        const int cbase = ch * CHUNK;
        // ---- stage codebook chunk (128 codes x 256 dims)
#if ASYNC_LDS
#pragma unroll
        for (int i = 0; i < (CHUNK * DIM / 4) / 256; ++i) {  // 32 iters
            int f   = i * 256 + tid;
            int row = f >> 6;
            int c4  = f & 63;
            const float* g = cb + (size_t)(cbase + row) * DIM + c4 * 4;
            float*       l = &Bs[row * LDA + c4 * 4];
            ASYNC_COPY_B128(g, l);
        }
        // warm next chunk into cache while the async transfer drains
        if (ch + 1 < NCHUNK) {
            const char* nb = (const char*)(cb + (size_t)(cbase + CHUNK) * DIM);
#pragma unroll
            for (int j = 0; j < 4; ++j)
                __builtin_prefetch(nb + tid * 512 + j * 128, 0, 0);
        }
        asm volatile("s_wait_asynccnt 0x0" ::: "memory");
#else
        const float4* cb4 = (const float4*)cb;
#pragma unroll
        for (int i = 0; i < (CHUNK * DIM / 4) / 256; ++i) {  // 32 iters
            int f   = i * 256 + tid;
            int row = f >> 6;
            int c4  = f & 63;
            float4 v = cb4[(size_t)(cbase + row) * (DIM / 4) + c4];
            *(float4*)&Bs[row * LDA + c4 * 4] = v;
        }
#endif
        __syncthreads();

        // ---- 16x32 fp32 WMMA tile per wave: A fragment reused across two B tiles
        v8f acc0 = {0.0f, 0.0f, 0.0f, 0.0f, 0.0f, 0.0f, 0.0f, 0.0f};
        v8f acc1 = {0.0f, 0.0f, 0.0f, 0.0f, 0.0f, 0.0f, 0.0f, 0.0f};
#pragma unroll
        for (int kk = 0; kk < DIM; kk += 4) {
            v2f a  = *(const v2f*)(aRow  + kk + koff);
            v2f b0 = *(const v2f*)(bRow0 + kk + koff);
            v2f b1 = *(const v2f*)(bRow1 + kk + koff);
            acc0 = __builtin_amdgcn_wmma_f32_16x16x4_f32(
                false, a, false, b0, (short)0, acc0, false, false);
            acc1 = __builtin_amdgcn_wmma_f32_16x16x4_f32(
                false, a, false, b1, (short)0, acc1, false, false);
        }

        // ---- epilogue: rel dist = |c|^2 - 2 x.c ; C/D layout lanes0-15: M=r, lanes16-31: M=r+8
        const int ng0 = cbase + npair * 32 + lane16;        // global code ids
        const int ng1 = ng0 + 16;
        const float c20 = ws[WS_C2 + ng0];
        const float c21 = ws[WS_C2 + ng1];
        const int mb = mtile * 16 + ((lane < 16) ? 0 : 8);
#pragma unroll
        for (int r = 0; r < 8; ++r) {
            Ds[(mb + r) * LDD + ng0] = c20 - 2.0f * acc0[r];
            Ds[(mb + r) * LDD + ng1] = c21 - 2.0f * acc1[r];
        }
        __syncthreads();
    }

    // ---- phase 2: per-point argmin + online log-sum-exp (8 lanes per point, strided n)
    const int p = tid >> 3;           // 0..31 local point
    const int s = tid & 7;            // segment
    const float* drow = &Ds[p * LDD];

    float bestV = 3.4e38f; int bestI = 0;
    float zm = -1.0e30f, Ssum = 0.0f, Tsum = 0.0f;  // max, sum e^{z-zm}, sum (z-zm)e^{z-zm}
    for (int i = 0; i < KCODES / 8; ++i) {
        int   n = s + 8 * i;
        float d = drow[n];
        if (d < bestV || (d == bestV && n < bestI)) { bestV = d; bestI = n; }
        float z = -d * INV_T;
        if (z > zm) {
            float c = __expf(zm - z);
            Tsum = c * (Tsum + (zm - z) * Ssum);
            Ssum = c * Ssum + 1.0f;
            zm   = z;
        } else {
            float e = __expf(z - zm);
            Ssum += e;
            Tsum += (z - zm) * e;
        }
    }
    // merge the 8 segment partials (wave32 shuffles within group of 8)
#pragma unroll
    for (int o = 1; o < 8; o <<= 1) {
        float oV = __shfl_xor(bestV, o, 32);
        int   oI = __shfl_xor(bestI, o, 32);
        float oZ = __shfl_xor(zm,    o, 32);
        float oS = __shfl_xor(Ssum,  o, 32);
        float oT = __shfl_xor(Tsum,  o, 32);
        if (oV < bestV || (oV == bestV && oI < bestI)) { bestV = oV; bestI = oI; }
        if (oZ > zm) {
            float c = __expf(zm - oZ);
            Tsum = oT + c * (Tsum + (zm - oZ) * Ssum);
            Ssum = oS + c * Ssum;
            zm   = oZ;
        } else {
            float c = __expf(oZ - zm);
            Tsum = Tsum + c * (oT + (oZ - zm) * oS);
            Ssum = Ssum + c * oS;
        }
    }

    // avg_probs partial: p_k = e^{z-zm}/S  (ds_add_f32 atomics into LDS staging)
    const float invS = 1.0f / Ssum;
    for (int i = 0; i < KCODES / 8; ++i) {
        int   n = s + 8 * i;
        float z = -drow[n] * INV_T;
        atomicAdd(&Pp[n], __expf(z - zm) * invS);
    }
    if (s == 0) {
        out[QOFF + 1 + pbase + p] = (float)bestI;            // encoding index
        atomicAdd(&ws[WS_ENT], Tsum * invS - __logf(Ssum));  // sum_k p*logp for this point
    }
    __syncthreads();
    for (int j = tid; j < KCODES; j += 256)
        atomicAdd(&ws[WS_AVG + j], Pp[j]);
}

// ---------------------------------------------------------------- gather quantized + MSE
__global__ __launch_bounds__(256)
void vq_gather_kernel(const float* __restrict__ x, const float* __restrict__ cb,
                      float* __restrict__ out, float* __restrict__ ws) {
    __shared__ float red[256];
    int fe  = blockIdx.x * 256 + threadIdx.x;   // float4 element id
    int p   = fe >> 6;
    int d4  = fe & 63;
    int idx = (int)out[QOFF + 1 + p];
    float4 q  = ((const float4*)cb)[idx * (DIM / 4) + d4];
    float4 xv = ((const float4*)x)[fe];
    ((float4*)out)[fe] = q;                     // straight-through forward value == codebook row
    float dx = q.x - xv.x, dy = q.y - xv.y, dz = q.z - xv.z, dw = q.w - xv.w;
    red[threadIdx.x] = dx * dx + dy * dy + dz * dz + dw * dw;
    __syncthreads();
    for (int o = 128; o > 0; o >>= 1) {
        if (threadIdx.x < o) red[threadIdx.x] += red[threadIdx.x + o];
        __syncthreads();
    }
    if (threadIdx.x == 0) atomicAdd(&ws[WS_MSE], red[0]);
}

// ---------------------------------------------------------------- final scalar loss
__global__ __launch_bounds__(256)
void vq_loss_kernel(const float* __restrict__ ws, float* __restrict__ out) {
    __shared__ float red[256];
    float acc = 0.0f;
    for (int j = threadIdx.x; j < KCODES; j += 256) {
        float ap = ws[WS_AVG + j] * (1.0f / (float)NPTS);
        acc += -ap * __logf(ap + 1e-5f);
    }
    red[threadIdx.x] = acc;
    __syncthreads();
    for (int o = 128; o > 0; o >>= 1) {
        if (threadIdx.x < o) red[threadIdx.x] += red[threadIdx.x + o];
        __syncthreads();
    }
    if (threadIdx.x == 0) {
        float avg_entropy    = red[0];
        float sample_entropy = -ws[WS_ENT] * (1.0f / (float)NPTS);
        float entropy_loss   = sample_entropy - avg_entropy;
        float mse            = ws[WS_MSE] * (1.0f / (float)(NPTS * DIM));
        out[QOFF] = 1.25f * mse + 0.1f * entropy_loss;   // e_latent(0.25x)+q_latent + 0.1*entropy
    }
}

// ----------------------------------------------------------------
extern "C" void kernel_launch(void* const* d_in, const int* in_sizes, int n_in,
                              void* d_out, int out_size, void* d_ws, size_t ws_size,
                              hipStream_t stream) {
    const float* x  = (const float*)d_in[0];   // (64,32,32,256) fp32
    const float* cb = (const float*)d_in[1];   // (1024,256) fp32
    float* out = (float*)d_out;
    float* ws  = (float*)d_ws;

    vq_zero_kernel<<<1, 256, 0, stream>>>(ws);
    vq_c2_kernel<<<(KCODES + 255) / 256, 256, 0, stream>>>(cb, ws);

    size_t smem = (size_t)(BM * LDA + CHUNK * LDA + BM * LDD + KCODES) * sizeof(float);
    vq_dist_kernel<<<NPTS / BM, 256, smem, stream>>>(x, cb, out, ws);

    vq_gather_kernel<<<(NPTS * DIM / 4) / 256, 256, 0, stream>>>(x, cb, out, ws);
    vq_loss_kernel<<<1, 256, 0, stream>>>(ws, out);
}